// DataAugmentation_84499186582018
// MI455X (gfx1250) — compile-verified
//
#include <hip/hip_runtime.h>
#include <math.h>

// ---------------------------------------------------------------------------
// DataAugmentation on MI455X (gfx1250, wave32).
//  Stage 1: fused (x + 0.45*noise, *1.1, clip) -> perspective bilinear ->
//           rotation bilinear, restricted to the 160x160 crop window.
//  Stage 2: bicubic 160->224 resize expressed as two f32 GEMMs on the WMMA
//           pipe:  out = Wv (224x160) @ img (160x160) @ Ww^T (160x224),
//           with Wv == Ww (same axis resize). Uses V_WMMA_F32_16X16X4_F32.
// ---------------------------------------------------------------------------

typedef __attribute__((ext_vector_type(2))) float v2f;
typedef __attribute__((ext_vector_type(8))) float v8f;

#define HH 224
#define WW 224
#define CROP 160
#define NIMG 384            // 128 * 3 channel-images
#define CROP_OFF 32

struct WarpParams {
  float pa, pb, pc, pd, pe, pf, pg, ph;   // perspective coeffs
  float ccos, csin;                       // rotation
};

// ---------------- Stage 1: fused elementwise + double bilinear warp --------

__device__ __forceinline__ float pre_tap(const float* __restrict__ xp,
                                         const float* __restrict__ np_,
                                         int yi, int xi) {
  bool valid = (xi >= 0) & (xi <= WW - 1) & (yi >= 0) & (yi <= HH - 1);
  int xc = xi < 0 ? 0 : (xi > WW - 1 ? WW - 1 : xi);
  int yc = yi < 0 ? 0 : (yi > HH - 1 ? HH - 1 : yi);
  int o = yc * WW + xc;
  float v = xp[o] + 0.45f * np_[o];
  v = v * 1.1f;
  v = fminf(fmaxf(v, 0.0f), 1.0f);
  return valid ? v : 0.0f;
}

// value of the perspective-warped, preprocessed image at integer (xi, yi)
__device__ __forceinline__ float persp_val(const float* __restrict__ xp,
                                           const float* __restrict__ np_,
                                           float xi, float yi, WarpParams P) {
  float denom = P.pg * xi + P.ph * yi + 1.0f;
  float inv = 1.0f / denom;
  float sx = (P.pa * xi + P.pb * yi + P.pc) * inv;
  float sy = (P.pd * xi + P.pe * yi + P.pf) * inv;
  float x0f = floorf(sx), y0f = floorf(sy);
  int   x0 = (int)x0f,   y0 = (int)y0f;
  float wx1 = sx - x0f, wx0 = 1.0f - wx1;
  float wy1 = sy - y0f, wy0 = 1.0f - wy1;
  return pre_tap(xp, np_, y0,     x0    ) * (wy0 * wx0)
       + pre_tap(xp, np_, y0,     x0 + 1) * (wy0 * wx1)
       + pre_tap(xp, np_, y0 + 1, x0    ) * (wy1 * wx0)
       + pre_tap(xp, np_, y0 + 1, x0 + 1) * (wy1 * wx1);
}

__global__ void warp_kernel(const float* __restrict__ x,
                            const float* __restrict__ nz,
                            float* __restrict__ rot, WarpParams P) {
  int idx = blockIdx.x * blockDim.x + threadIdx.x;     // NIMG*160*160 exactly
  int bc = idx / (CROP * CROP);
  int r  = idx - bc * (CROP * CROP);
  int yy = r / CROP;
  int xx = r - yy * CROP;

  const float* xp  = x  + (size_t)bc * HH * WW;
  const float* np_ = nz + (size_t)bc * HH * WW;

  float xo = (float)(xx + CROP_OFF);
  float yo = (float)(yy + CROP_OFF);
  const float cx = (WW - 1) * 0.5f, cy = (HH - 1) * 0.5f;
  float dx = xo - cx, dy = yo - cy;
  float rx = cx + P.ccos * dx + P.csin * dy;
  float ry = cy - P.csin * dx + P.ccos * dy;

  float x0f = floorf(rx), y0f = floorf(ry);
  int   x0 = (int)x0f,   y0 = (int)y0f;
  float wx1 = rx - x0f, wx0 = 1.0f - wx1;
  float wy1 = ry - y0f, wy0 = 1.0f - wy1;

  float acc = 0.0f;
  if (x0     >= 0 && x0     < WW && y0     >= 0 && y0     < HH)
    acc += persp_val(xp, np_, (float)x0,       (float)y0      , P) * (wy0 * wx0);
  if (x0 + 1 >= 0 && x0 + 1 < WW && y0     >= 0 && y0     < HH)
    acc += persp_val(xp, np_, (float)(x0 + 1), (float)y0      , P) * (wy0 * wx1);
  if (x0     >= 0 && x0     < WW && y0 + 1 >= 0 && y0 + 1 < HH)
    acc += persp_val(xp, np_, (float)x0,       (float)(y0 + 1), P) * (wy1 * wx0);
  if (x0 + 1 >= 0 && x0 + 1 < WW && y0 + 1 >= 0 && y0 + 1 < HH)
    acc += persp_val(xp, np_, (float)(x0 + 1), (float)(y0 + 1), P) * (wy1 * wx1);

  rot[idx] = acc;
}

// ---------------- Stage 2a: bicubic weight matrix (224 x 160) --------------

__device__ __forceinline__ float keys_cubic(float x) {   // a = -0.5 (Keys)
  x = fabsf(x);
  if (x < 1.0f) return ((1.5f * x - 2.5f) * x) * x + 1.0f;
  if (x < 2.0f) return ((-0.5f * x + 2.5f) * x - 4.0f) * x + 2.0f;
  return 0.0f;
}

__global__ void weights_kernel(float* __restrict__ Wgt) {
  int i = blockIdx.x * blockDim.x + threadIdx.x;   // output row, 0..223
  if (i >= HH) return;
  float sf = (i + 0.5f) * ((float)CROP / (float)HH) - 0.5f;  // sample pos
  float* row = Wgt + i * CROP;
  float sum = 0.0f;
  for (int j = 0; j < CROP; ++j) {
    float w = keys_cubic(sf - (float)j);
    row[j] = w;
    sum += w;
  }
  float inv = 1.0f / sum;                          // jax normalizes columns
  for (int j = 0; j < CROP; ++j) row[j] *= inv;
}

// ---------------- Stage 2b/2c: WMMA f32 GEMMs ------------------------------
// V_WMMA_F32_16X16X4_F32 operand layout (ISA 7.12.2, wave32):
//   A (16x4, MxK): lane 0-15 -> M=lane, VGPR0=K0,VGPR1=K1;
//                  lane16-31 -> M=lane-16, VGPR0=K2,VGPR1=K3.
//   B (4x16, KxN): lane selects N=lane%16; half-wave selects K pair.
//   C/D (16x16):   VGPR r, lane<16 -> M=r, N=lane; lane>=16 -> M=r+8.

__device__ __forceinline__ v8f wmma_f32(v2f a, v2f b, v8f c) {
  return __builtin_amdgcn_wmma_f32_16x16x4_f32(
      /*neg_a=*/false, a, /*neg_b=*/false, b,
      /*c_mod=*/(short)0, c, /*reuse_a=*/false, /*reuse_b=*/false);
}

// tmp[bc][i][x] = sum_y Wgt[i][y] * rot[bc][y][x]    (224x160 = 224x160 @ 160x160)
__global__ void gemm1_kernel(const float* __restrict__ Wgt,
                             const float* __restrict__ rot,
                             float* __restrict__ tmp) {
  const int Nt = CROP / 16;            // 10 tiles in N
  const int tilesPerImg = (HH / 16) * Nt;   // 140
  int wave = (blockIdx.x * blockDim.x + threadIdx.x) >> 5;
  int lane = threadIdx.x & 31;
  int bc = wave / tilesPerImg;
  int t  = wave - bc * tilesPerImg;
  int i0 = (t / Nt) * 16;
  int j0 = (t - (t / Nt) * Nt) * 16;
  int half = lane >> 4;                // K-pair selector
  int l16  = lane & 15;

  const float* Bm = rot + (size_t)bc * CROP * CROP;
  const float* arow = Wgt + (i0 + l16) * CROP;
  v8f acc = {};
  for (int k = 0; k < CROP; k += 4) {
    v2f a = *(const v2f*)(arow + k + 2 * half);          // A[M][K..K+1]
    const float* bp = Bm + (k + 2 * half) * CROP + j0 + l16;
    v2f b; b.x = bp[0]; b.y = bp[CROP];                  // B[K][N], B[K+1][N]
    acc = wmma_f32(a, b, acc);
  }
  float* drow = tmp + (size_t)bc * HH * CROP + (i0 + 8 * half) * CROP + j0 + l16;
#pragma unroll
  for (int r = 0; r < 8; ++r) drow[r * CROP] = acc[r];
}

// out[bc][i][j] = sum_x tmp[bc][i][x] * Wgt[j][x]      (224x224 = 224x160 @ 160x224)
__global__ void gemm2_kernel(const float* __restrict__ tmp,
                             const float* __restrict__ Wgt,
                             float* __restrict__ out) {
  const int Nt = WW / 16;              // 14
  const int tilesPerImg = (HH / 16) * Nt;   // 196
  int wave = (blockIdx.x * blockDim.x + threadIdx.x) >> 5;
  int lane = threadIdx.x & 31;
  int bc = wave / tilesPerImg;
  int t  = wave - bc * tilesPerImg;
  int i0 = (t / Nt) * 16;
  int j0 = (t - (t / Nt) * Nt) * 16;
  int half = lane >> 4;
  int l16  = lane & 15;

  const float* arow = tmp + (size_t)bc * HH * CROP + (i0 + l16) * CROP;
  const float* brow = Wgt + (j0 + l16) * CROP;     // B[K][N] = Wgt[N][K]
  v8f acc = {};
  for (int k = 0; k < CROP; k += 4) {
    v2f a = *(const v2f*)(arow + k + 2 * half);
    v2f b = *(const v2f*)(brow + k + 2 * half);
    acc = wmma_f32(a, b, acc);
  }
  float* drow = out + (size_t)bc * HH * WW + (i0 + 8 * half) * WW + j0 + l16;
#pragma unroll
  for (int r = 0; r < 8; ++r) drow[r * WW] = acc[r];
}

// ---------------- host side -----------------------------------------------

static void solve8x8(double A[8][8], double b[8], double out[8]) {
  for (int col = 0; col < 8; ++col) {
    int piv = col;
    for (int r2 = col + 1; r2 < 8; ++r2)
      if (fabs(A[r2][col]) > fabs(A[piv][col])) piv = r2;
    if (piv != col) {
      for (int c2 = 0; c2 < 8; ++c2) { double tt = A[col][c2]; A[col][c2] = A[piv][c2]; A[piv][c2] = tt; }
      double tb = b[col]; b[col] = b[piv]; b[piv] = tb;
    }
    double inv = 1.0 / A[col][col];
    for (int r2 = col + 1; r2 < 8; ++r2) {
      double f = A[r2][col] * inv;
      for (int c2 = col; c2 < 8; ++c2) A[r2][c2] -= f * A[col][c2];
      b[r2] -= f * b[col];
    }
  }
  for (int r2 = 7; r2 >= 0; --r2) {
    double s = b[r2];
    for (int c2 = r2 + 1; c2 < 8; ++c2) s -= A[r2][c2] * out[c2];
    out[r2] = s / A[r2][r2];
  }
}

extern "C" void kernel_launch(void* const* d_in, const int* in_sizes, int n_in,
                              void* d_out, int out_size, void* d_ws, size_t ws_size,
                              hipStream_t stream) {
  (void)in_sizes; (void)n_in; (void)out_size; (void)ws_size;
  const float* x  = (const float*)d_in[0];
  const float* nz = (const float*)d_in[1];
  float* out = (float*)d_out;

  char* ws = (char*)d_ws;
  // ws layout (all 256B aligned):
  float* rot = (float*)(ws);                               // 384*160*160 = 39,321,600 B
  float* Wgt = (float*)(ws + 39321600u);                   // 224*160*4   =    143,360 B
  float* tmp = (float*)(ws + 39464960u);                   // 384*224*160 = 55,050,240 B

  // perspective coefficients (same solve numpy does, in double)
  const double sp[4][2] = {{0, 0}, {223, 0}, {223, 223}, {0, 223}};
  const double ep[4][2] = {{30, 20}, {200, 25}, {210, 200}, {15, 190}};
  double A[8][8], b[8], co[8];
  for (int k = 0; k < 8; ++k) for (int c = 0; c < 8; ++c) A[k][c] = 0.0;
  for (int k = 0; k < 4; ++k) {
    double xi = sp[k][0], yi = sp[k][1], xo = ep[k][0], yo = ep[k][1];
    A[2 * k][0] = xo; A[2 * k][1] = yo; A[2 * k][2] = 1.0;
    A[2 * k][6] = -xo * xi; A[2 * k][7] = -yo * xi; b[2 * k] = xi;
    A[2 * k + 1][3] = xo; A[2 * k + 1][4] = yo; A[2 * k + 1][5] = 1.0;
    A[2 * k + 1][6] = -xo * yi; A[2 * k + 1][7] = -yo * yi; b[2 * k + 1] = yi;
  }
  solve8x8(A, b, co);

  WarpParams P;
  P.pa = (float)co[0]; P.pb = (float)co[1]; P.pc = (float)co[2]; P.pd = (float)co[3];
  P.pe = (float)co[4]; P.pf = (float)co[5]; P.pg = (float)co[6]; P.ph = (float)co[7];
  const double theta = 10.0 * 0.017453292519943295;
  P.ccos = (float)cos(theta); P.csin = (float)sin(theta);

  // 1) bicubic weight matrix (device-built; no H2D copies during capture)
  weights_kernel<<<7, 32, 0, stream>>>(Wgt);
  // 2) fused elementwise + perspective + rotation, crop window only
  warp_kernel<<<38400, 256, 0, stream>>>(x, nz, rot, P);     // 384*160*160 threads
  // 3) vertical resize: tmp = Wgt @ rot   (WMMA f32, one wave per 16x16 tile)
  gemm1_kernel<<<6720, 256, 0, stream>>>(Wgt, rot, tmp);     // 53,760 tiles / 8 waves
  // 4) horizontal resize: out = tmp @ Wgt^T
  gemm2_kernel<<<9408, 256, 0, stream>>>(tmp, Wgt, out);     // 75,264 tiles / 8 waves
}